// PairwiseModalityAttention_75445395521835
// MI455X (gfx1250) — compile-verified
//
#include <hip/hip_runtime.h>

// ---------------------------------------------------------------------------
// PairwiseModalityAttention for MI455X (gfx1250).
//
// Mathematical simplification: softmax over a size-1 key axis == 1, so the
// whole Q/K path is dead.  out_segment[i] = mask[:,i] * 0.5 * sum_{j!=i}
// ( feats[j] @ (Wo_ij @ Wv_ij)^T + (Wo_ij @ bv_ij + bo_ij) ).
//
// Stage 1: precompute Weff = 0.5*Wo@Wv per pair (k-major K x 256) + biases.
// Stage 2: three streaming f32 WMMA GEMMs (16x16x4_f32), memory-bound:
//          ~560MB min HBM traffic -> ~24us floor at 23.3 TB/s.
// ---------------------------------------------------------------------------

typedef __attribute__((ext_vector_type(2))) float v2f;
typedef __attribute__((ext_vector_type(8))) float v8f;

#define HID 256

// ---------------------------------------------------------------------------
// Stage 1a: Weff[d][n] = 0.5 * sum_c Wo[n][c] * Wv[c][d]   (k-major output)
// One block per 4 consecutive d; thread n computes 4 outputs.
// ---------------------------------------------------------------------------
__global__ void wseg_kernel(const float* __restrict__ Wv,   // (256, kd) row-major
                            const float* __restrict__ Wo,   // (256, 256) row-major
                            int kd,
                            float* __restrict__ outW)       // (kd, 256) k-major
{
    __shared__ float vcol[4][HID];
    const int t  = threadIdx.x;        // 0..255
    const int d4 = blockIdx.x * 4;
    for (int q = 0; q < 4; ++q)
        vcol[q][t] = Wv[(size_t)t * kd + d4 + q];
    __syncthreads();

    float a0 = 0.f, a1 = 0.f, a2 = 0.f, a3 = 0.f;
    const float* wrow = Wo + (size_t)t * HID;
    for (int c = 0; c < HID; ++c) {
        float w = wrow[c];
        a0 += w * vcol[0][c];
        a1 += w * vcol[1][c];
        a2 += w * vcol[2][c];
        a3 += w * vcol[3][c];
    }
    float* o = outW + (size_t)d4 * HID + t;
    o[0]       = 0.5f * a0;
    o[HID]     = 0.5f * a1;
    o[2 * HID] = 0.5f * a2;
    o[3 * HID] = 0.5f * a3;
}

// ---------------------------------------------------------------------------
// Stage 1b: per-segment bias = 0.5 * sum over the segment's two pairs of
//           (Wo @ bv + bo).
// ---------------------------------------------------------------------------
__global__ void bias_kernel(const float* __restrict__ Wo1, const float* __restrict__ bv1,
                            const float* __restrict__ bo1,
                            const float* __restrict__ Wo2, const float* __restrict__ bv2,
                            const float* __restrict__ bo2,
                            float* __restrict__ outB)
{
    const int n = threadIdx.x;
    float a = bo1[n];
    float b = bo2[n];
    for (int c = 0; c < HID; ++c) {
        a += Wo1[(size_t)n * HID + c] * bv1[c];
        b += Wo2[(size_t)n * HID + c] * bv2[c];
    }
    outB[n] = 0.5f * (a + b);
}

// ---------------------------------------------------------------------------
// Stage 2: fused streaming GEMM for one output segment.
//   grid.x = 4 column groups (64 cols each), grid.y = B/128 row groups.
//   Block = 256 threads = 8 waves; wave w owns one 16-row WMMA tile.
//   Each wave accumulates 4 adjacent 16x16 output tiles (64 cols) in v8f regs.
//   Weights (K x 64 slice of Weff) streamed through LDS in 128-deep K chunks,
//   stored pre-swizzled as B fragments:
//     ldsB[(k>>1)*128 + n*2 + (k&1)] = W[k][n]
//   so a lane's B fragment is one 8-byte LDS load.
//
//   WMMA f32 16x16x4 layouts (ISA 7.12.2):
//     A: lane l (l&15 = M row), VGPR0/1 = K {0,1} (lanes<16) or {2,3} (lanes>=16)
//     B: lane l (l&15 = N col), VGPR0/1 = K {0,1} / {2,3}  (transpose of A)
//     C: lane l (l&15 = N), vgpr r = row M=r (lanes<16) or M=r+8 (lanes>=16)
// ---------------------------------------------------------------------------
__global__ __launch_bounds__(256)
void fused_gemm(const float* __restrict__ p0, int d0,   // first  input (B, d0)
                const float* __restrict__ p1, int d1,   // second input (B, d1)
                const int*   __restrict__ mask,         // (B, 3)
                const float* __restrict__ wseg,         // (d0+d1, 256) k-major
                const float* __restrict__ bias,         // (256,)
                float*       __restrict__ out,          // (B, 768)
                int seg)
{
    __shared__ float ldsB[128 / 2 * 128];   // 8192 floats = 32 KB

    const int tid     = threadIdx.x;
    const int lane    = tid & 31;
    const int wave    = tid >> 5;
    const int ln      = lane & 15;          // M row (A) / N col (B,C)
    const int hi      = lane >> 4;          // 0: K0-1 / rows 0-7, 1: K2-3 / rows 8-15
    const int nbase   = blockIdx.x * 64;
    const int rowbase = (blockIdx.y * 8 + wave) * 16;

    v8f acc0 = {0.f,0.f,0.f,0.f,0.f,0.f,0.f,0.f};
    v8f acc1 = acc0, acc2 = acc0, acc3 = acc0;

    const float* parts[2] = {p0, p1};
    const int    dims[2]  = {d0, d1};

    int kglob = 0;
    for (int part = 0; part < 2; ++part) {
        const float* xrow = parts[part] + (size_t)(rowbase + ln) * dims[part];
        const int dim = dims[part];
        for (int kc = 0; kc < dim; kc += 128) {
            const int clen = (dim - kc) < 128 ? (dim - kc) : 128;   // 64 or 128

            __syncthreads();
            // Stage the (clen x 64) weight slice, swizzled into B-fragment form.
            for (int idx = tid; idx < clen * 64; idx += 256) {
                const int kk = idx >> 6;
                const int n  = idx & 63;
                ldsB[((kk >> 1) << 7) + (n << 1) + (kk & 1)] =
                    wseg[(size_t)(kglob + kk) * HID + nbase + n];
            }
            __syncthreads();

            for (int kk = 0; kk < clen; kk += 4) {
                // A fragment: 2 consecutive f32 of this lane's row at k + {0,2}
                v2f a = *(const v2f*)(xrow + kc + kk + hi * 2);
                // B fragments for the 4 N tiles
                const float* bp = &ldsB[(((kk >> 1) + hi) << 7) + (ln << 1)];
                v2f b0 = *(const v2f*)(bp);
                v2f b1 = *(const v2f*)(bp + 32);
                v2f b2 = *(const v2f*)(bp + 64);
                v2f b3 = *(const v2f*)(bp + 96);

                acc0 = __builtin_amdgcn_wmma_f32_16x16x4_f32(false, a, false, b0,
                                                             (short)0, acc0, false, false);
                acc1 = __builtin_amdgcn_wmma_f32_16x16x4_f32(false, a, false, b1,
                                                             (short)0, acc1, false, false);
                acc2 = __builtin_amdgcn_wmma_f32_16x16x4_f32(false, a, false, b2,
                                                             (short)0, acc2, false, false);
                acc3 = __builtin_amdgcn_wmma_f32_16x16x4_f32(false, a, false, b3,
                                                             (short)0, acc3, false, false);
            }
            kglob += clen;
        }
    }

    // Epilogue: out[row, seg*256 + nbase + n] = mask[row,seg] * (acc + bias)
    const float b0v = bias[nbase + ln];
    const float b1v = bias[nbase + 16 + ln];
    const float b2v = bias[nbase + 32 + ln];
    const float b3v = bias[nbase + 48 + ln];
    const size_t colbase = (size_t)seg * 256 + nbase + ln;

#pragma unroll
    for (int r = 0; r < 8; ++r) {
        const int row = rowbase + r + hi * 8;
        const float s = (mask[(size_t)row * 3 + seg] != 0) ? 1.0f : 0.0f;
        float* o = out + (size_t)row * 768 + colbase;
        o[0]  = (acc0[r] + b0v) * s;
        o[16] = (acc1[r] + b1v) * s;
        o[32] = (acc2[r] + b2v) * s;
        o[48] = (acc3[r] + b3v) * s;
    }
}

// ---------------------------------------------------------------------------
// Launcher.  Input flattening (recursive dict insertion order):
//   0: video (B,512)  1: audio (B,128)  2: imu (B,64)  3: mask (B,3) int32
//   then 6 pairs in _pairs() order, 8 tensors each: Wq bq Wk bk Wv bv Wo bo
// ---------------------------------------------------------------------------
extern "C" void kernel_launch(void* const* d_in, const int* in_sizes, int n_in,
                              void* d_out, int out_size, void* d_ws, size_t ws_size,
                              hipStream_t stream)
{
    const float* video = (const float*)d_in[0];
    const float* audio = (const float*)d_in[1];
    const float* imu   = (const float*)d_in[2];
    const int*   mask  = (const int*)d_in[3];
    const int Bn = in_sizes[0] / 512;          // 65536

    // Pair parameter bases (insertion order): v2a, v2i, a2v, a2i, i2v, i2a
    const int PB_V2A = 4, PB_V2I = 12, PB_A2V = 20, PB_A2I = 28, PB_I2V = 36, PB_I2A = 44;
    auto Wv = [&](int b) { return (const float*)d_in[b + 4]; };
    auto bv = [&](int b) { return (const float*)d_in[b + 5]; };
    auto Wo = [&](int b) { return (const float*)d_in[b + 6]; };
    auto bo = [&](int b) { return (const float*)d_in[b + 7]; };

    // Workspace: Wseg0 (192x256) | Wseg1 (576x256) | Wseg2 (640x256) | bias (3x256)
    float* wsF   = (float*)d_ws;
    float* wseg0 = wsF;
    float* wseg1 = wsF + 192 * HID;
    float* wseg2 = wsF + (192 + 576) * HID;
    float* biasB = wsF + (192 + 576 + 640) * HID;

    // Stage 1a: effective weights (k-major), per segment's two K parts.
    // seg0 (video out): K = [audio 128 | imu 64], pairs v2a, v2i
    wseg_kernel<<<128 / 4, 256, 0, stream>>>(Wv(PB_V2A), Wo(PB_V2A), 128, wseg0);
    wseg_kernel<<< 64 / 4, 256, 0, stream>>>(Wv(PB_V2I), Wo(PB_V2I),  64, wseg0 + 128 * HID);
    // seg1 (audio out): K = [video 512 | imu 64], pairs a2v, a2i
    wseg_kernel<<<512 / 4, 256, 0, stream>>>(Wv(PB_A2V), Wo(PB_A2V), 512, wseg1);
    wseg_kernel<<< 64 / 4, 256, 0, stream>>>(Wv(PB_A2I), Wo(PB_A2I),  64, wseg1 + 512 * HID);
    // seg2 (imu out): K = [video 512 | audio 128], pairs i2v, i2a
    wseg_kernel<<<512 / 4, 256, 0, stream>>>(Wv(PB_I2V), Wo(PB_I2V), 512, wseg2);
    wseg_kernel<<<128 / 4, 256, 0, stream>>>(Wv(PB_I2A), Wo(PB_I2A), 128, wseg2 + 512 * HID);

    // Stage 1b: per-segment biases.
    bias_kernel<<<1, 256, 0, stream>>>(Wo(PB_V2A), bv(PB_V2A), bo(PB_V2A),
                                       Wo(PB_V2I), bv(PB_V2I), bo(PB_V2I), biasB);
    bias_kernel<<<1, 256, 0, stream>>>(Wo(PB_A2V), bv(PB_A2V), bo(PB_A2V),
                                       Wo(PB_A2I), bv(PB_A2I), bo(PB_A2I), biasB + 256);
    bias_kernel<<<1, 256, 0, stream>>>(Wo(PB_I2V), bv(PB_I2V), bo(PB_I2V),
                                       Wo(PB_I2A), bv(PB_I2A), bo(PB_I2A), biasB + 512);

    // Stage 2: three fused WMMA GEMMs (one per output segment).
    dim3 grid(4, Bn / 128);
    float* out = (float*)d_out;
    fused_gemm<<<grid, 256, 0, stream>>>(audio, 128, imu,    64, mask, wseg0, biasB,       out, 0);
    fused_gemm<<<grid, 256, 0, stream>>>(video, 512, imu,    64, mask, wseg1, biasB + 256, out, 1);
    fused_gemm<<<grid, 256, 0, stream>>>(video, 512, audio, 128, mask, wseg2, biasB + 512, out, 2);
}